// Custom_Model_Group_Embedding_Bag_Addmm_1dbias_80625126081362
// MI455X (gfx1250) — compile-verified
//
#include <hip/hip_runtime.h>
#include <hip/hip_bf16.h>

typedef __attribute__((ext_vector_type(2)))  float v2f;
typedef __attribute__((ext_vector_type(8)))  float v8f;

#define WAVE 32

// -------------------------------------------------------------------------
// Kernel 1: EmbeddingBag(mean), one wave32 per bag.
// Lanes stride over the bag's indices, gather 3 floats from the (L2-resident)
// 120MB table, butterfly-reduce across the wave, lane0 writes the mean into
// output columns 0..8 (the three aliased EmbeddingBags).
// -------------------------------------------------------------------------
__global__ void emb_bag_mean_kernel(const int* __restrict__ eb_input,
                                    const int* __restrict__ eb_offset,
                                    const float* __restrict__ emb_weight,
                                    float* __restrict__ out,
                                    int B, int N_total) {
    const int wave_in_block = threadIdx.x >> 5;
    const int lane          = threadIdx.x & 31;
    const int bag = blockIdx.x * (blockDim.x >> 5) + wave_in_block;
    if (bag >= B) return;

    const int start = eb_offset[bag];
    const int end   = (bag + 1 < B) ? eb_offset[bag + 1] : N_total;

    float a0 = 0.f, a1 = 0.f, a2 = 0.f;
    for (int i = start + lane; i < end; i += WAVE) {
        const long long idx = (long long)eb_input[i];
        const float* row = emb_weight + idx * 3;
        a0 += row[0];
        a1 += row[1];
        a2 += row[2];
    }
    // wave32 butterfly reduction (CDNA5 is wave32-only; width must be 32)
    #pragma unroll
    for (int m = 16; m >= 1; m >>= 1) {
        a0 += __shfl_xor(a0, m, WAVE);
        a1 += __shfl_xor(a1, m, WAVE);
        a2 += __shfl_xor(a2, m, WAVE);
    }
    if (lane == 0) {
        const int cnt = end - start;
        const float inv = 1.0f / (float)(cnt > 0 ? cnt : 1);
        const float e0 = a0 * inv, e1 = a1 * inv, e2 = a2 * inv;
        float* o = out + (size_t)bag * 12;
        o[0] = e0; o[1] = e1; o[2] = e2;   // eb_out copy 0
        o[3] = e0; o[4] = e1; o[5] = e2;   // eb_out copy 1
        o[6] = e0; o[7] = e1; o[8] = e2;   // eb_out copy 2
    }
}

// -------------------------------------------------------------------------
// Kernel 2: MLP. One wave per 16-row tile.
// Layer0 (128->12) via V_WMMA_F32_16X16X4_F32: w0^T is staged zero-padded
// (16x128) into LDS once per block, so the K-loop is a branch-free
// global_load_b64 (A) + ds_load_b64 (B) + v_wmma chain with fp32 accumulate.
// Tile dumped to LDS in C/D layout; lanes 0..15 run the tiny 12->6->3 tail
// and write output columns 9..11.
// -------------------------------------------------------------------------
__global__ void mlp_wmma_kernel(const float* __restrict__ x,   // [B,128]
                                const float* __restrict__ w0,  // [12,128]
                                const float* __restrict__ b0,  // [12]
                                const float* __restrict__ w1,  // [6,12]
                                const float* __restrict__ b1,  // [6]
                                const float* __restrict__ w2,  // [3,6]
                                const float* __restrict__ b2,  // [3]
                                float* __restrict__ out,       // [B,12]
                                int B) {
    __shared__ float w0pad[16 * 128];           // zero-padded w0 (rows 12..15 = 0)
    __shared__ float tilebuf[8 * 16 * 16];      // 8 waves/block * 16x16 D tile

    const int tid  = threadIdx.x;
    const int wave = tid >> 5;
    const int lane = tid & 31;
    const int half = lane >> 4;                 // K-group select within wave
    const int lm   = lane & 15;                 // row (A) / col (B,C,D) index

    // Stage zero-padded w0^T source once per block (16*128 floats, 256 thr).
    #pragma unroll
    for (int i = tid; i < 16 * 128; i += 256) {
        const int n = i >> 7;                   // 0..15
        const int k = i & 127;                  // 0..127
        w0pad[i] = (n < 12) ? w0[n * 128 + k] : 0.0f;
    }
    __syncthreads();

    const int tile   = blockIdx.x * (blockDim.x >> 5) + wave;
    const int ntiles = (B + 15) >> 4;
    const bool active = (tile < ntiles);        // wave-uniform (EXEC all-1s)
    float* my = &tilebuf[wave * 256];

    if (active) {
        const int row0 = tile * 16;
        v8f acc = {};

        // f32 A 16x4 layout: lane holds row lm, K = 2*half + {0,1} -> float2
        const float* arow = x + (size_t)(row0 + lm) * 128 + 2 * half;
        const float* brow = &w0pad[lm * 128 + 2 * half];   // B = padded w0^T
        #pragma unroll
        for (int s = 0; s < 32; ++s) {
            const int k0 = 4 * s;
            v2f a; a[0] = arow[k0]; a[1] = arow[k0 + 1];
            v2f b; b[0] = brow[k0]; b[1] = brow[k0 + 1];
            acc = __builtin_amdgcn_wmma_f32_16x16x4_f32(
                false, a, false, b, (short)0, acc, false, false);
        }

        // C/D layout: VGPR r holds M = r + 8*half, N = lm
        #pragma unroll
        for (int r = 0; r < 8; ++r) {
            my[(r + 8 * half) * 16 + lm] = acc[r];
        }
    }

    __syncthreads();   // outside 'active' guard: every wave reaches it

    if (active && lane < 16) {
        const int row = tile * 16 + lane;
        float h0[12];
        #pragma unroll
        for (int n = 0; n < 12; ++n) h0[n] = my[lane * 16 + n] + b0[n];
        float h1[6];
        #pragma unroll
        for (int j = 0; j < 6; ++j) {
            float s = b1[j];
            #pragma unroll
            for (int n = 0; n < 12; ++n) s += w1[j * 12 + n] * h0[n];
            h1[j] = s;
        }
        float* o = out + (size_t)row * 12 + 9;
        #pragma unroll
        for (int t = 0; t < 3; ++t) {
            float s = b2[t];
            #pragma unroll
            for (int j = 0; j < 6; ++j) s += w2[t * 6 + j] * h1[j];
            o[t] = s;
        }
    }
}

extern "C" void kernel_launch(void* const* d_in, const int* in_sizes, int n_in,
                              void* d_out, int out_size, void* d_ws, size_t ws_size,
                              hipStream_t stream) {
    const int*   eb_input   = (const int*)d_in[0];
    const int*   eb_offset  = (const int*)d_in[1];
    const float* mlp_input  = (const float*)d_in[2];
    const float* emb_weight = (const float*)d_in[3];
    const float* w0 = (const float*)d_in[4];
    const float* b0 = (const float*)d_in[5];
    const float* w1 = (const float*)d_in[6];
    const float* b1 = (const float*)d_in[7];
    const float* w2 = (const float*)d_in[8];
    const float* b2 = (const float*)d_in[9];
    float* out = (float*)d_out;

    const int B    = in_sizes[1];   // number of bags / batch
    const int Ntot = in_sizes[0];   // total indices (B*HIST)

    // One wave per bag, 8 waves (256 threads) per block.
    const int blocks1 = (B + 7) / 8;
    emb_bag_mean_kernel<<<blocks1, 256, 0, stream>>>(
        eb_input, eb_offset, emb_weight, out, B, Ntot);

    // One wave per 16-row tile, 8 waves per block.
    const int ntiles  = (B + 15) / 16;
    const int blocks2 = (ntiles + 7) / 8;
    mlp_wmma_kernel<<<blocks2, 256, 0, stream>>>(
        mlp_input, w0, b0, w1, b1, w2, b2, out, B);
}